// KernelDensity_10969346474407
// MI455X (gfx1250) — compile-verified
//
#include <hip/hip_runtime.h>
#include <math.h>

// CDNA5 / gfx1250 WMMA fragment types
typedef float v2f __attribute__((ext_vector_type(2)));
typedef float v8f __attribute__((ext_vector_type(8)));

#define DIMS 16
#define BLOCK_THREADS 512
#define WAVES (BLOCK_THREADS / 32)

// ---------------------------------------------------------------------------
// Prep: weff[n] = w[n] * exp(-0.5*|data_n|^2)   (one thread per data point)
// ---------------------------------------------------------------------------
__global__ __launch_bounds__(256)
void kde_prep_weff(const float* __restrict__ data,
                   const float* __restrict__ w,
                   float* __restrict__ weff,
                   int N)
{
    const int i = blockIdx.x * blockDim.x + threadIdx.x;
    if (i >= N) return;
    const float4* __restrict__ p = (const float4*)(data + (size_t)i * DIMS);
    float s = 0.0f;
#pragma unroll
    for (int q = 0; q < DIMS / 4; ++q) {
        float4 v = p[q];
        s += v.x * v.x + v.y * v.y + v.z * v.z + v.w * v.w;
    }
    weff[i] = w[i] * __expf(-0.5f * s);
}

// ---------------------------------------------------------------------------
// Shared epilogue: per-wave column reduce -> LDS -> log-density
// ---------------------------------------------------------------------------
__device__ __forceinline__
void kde_epilogue(float* lds, v8f& acc, const float* __restrict__ X,
                  float* __restrict__ out, int Q, int qbase,
                  int tid, int wave, int l16, int hi, float sum_w)
{
#pragma unroll
    for (int v = 0; v < 8; ++v) {
        float s = acc[v];
        s += __shfl_xor(s, 1, 32);
        s += __shfl_xor(s, 2, 32);
        s += __shfl_xor(s, 4, 32);
        s += __shfl_xor(s, 8, 32);
        acc[v] = s;                     // lane 0 -> m=v ; lane 16 -> m=8+v
    }
    if (l16 == 0) {
#pragma unroll
        for (int v = 0; v < 8; ++v)
            lds[wave * 16 + hi * 8 + v] = acc[v];
    }
    __syncthreads();

    if (tid < 16) {
        float s = 0.0f;
#pragma unroll
        for (int wv = 0; wv < WAVES; ++wv) s += lds[wv * 16 + tid];

        const int mq = qbase + tid;
        if (mq < Q) {
            const float* __restrict__ xr = X + (size_t)mq * DIMS;
            float xn2 = 0.0f;
#pragma unroll
            for (int k = 0; k < DIMS; ++k) xn2 += xr[k] * xr[k];
            const float LOG_2PI = 1.8378770664093453f;
            const float logc = -0.5f * (float)DIMS * LOG_2PI - logf(sum_w);
            out[mq] = logf(s) - 0.5f * xn2 + logc;
        }
    }
}

// ---------------------------------------------------------------------------
// Fast path: weights pre-folded with exp(-0.5*|d_n|^2); hot loop is
// 5 loads + 4 WMMA + 8 exp + 4 pk_fma, no DS ops, no EXEC juggling.
// ---------------------------------------------------------------------------
__global__ __launch_bounds__(BLOCK_THREADS)
void kde_wmma_fast(const float* __restrict__ X,
                   const float* __restrict__ data,
                   const float* __restrict__ weff,
                   const float* __restrict__ w,
                   float* __restrict__ out,
                   int Q, int N)
{
    __shared__ float lds[BLOCK_THREADS];

    const int tid  = threadIdx.x;
    const int wave = tid >> 5;          // wave32
    const int lane = tid & 31;
    const int l16  = lane & 15;
    const int hi   = lane >> 4;

    // block-wide sum of raw weights (L2-resident, cheap)
    float sw = 0.0f;
    for (int i = tid; i < N; i += BLOCK_THREADS) sw += w[i];
    lds[tid] = sw;
    __syncthreads();
    for (int s = BLOCK_THREADS / 2; s > 0; s >>= 1) {
        if (tid < s) lds[tid] += lds[tid + s];
        __syncthreads();
    }
    const float sum_w = lds[0];
    __syncthreads();

    // A fragments (ISA 32-bit 16x4 A layout): m = lane%16, k = 4c + 2*hi + r
    const int qbase = blockIdx.x * 16;
    int m = qbase + l16;
    if (m >= Q) m = Q - 1;
    const float* __restrict__ xrow = X + (size_t)m * DIMS;
    v2f afrag[4];
#pragma unroll
    for (int c = 0; c < 4; ++c)
        afrag[c] = *(const v2f*)(xrow + 4 * c + 2 * hi);

    v8f acc = {};
    const int nfull = N >> 4;           // full 16-point chunks (no guards)

    for (int j = wave; j < nfull; j += WAVES) {
        const int n = j * 16 + l16;
        const float* __restrict__ drow = data + (size_t)n * DIMS;
        const float wt = weff[n];

        v2f bfrag[4];
#pragma unroll
        for (int c = 0; c < 4; ++c)
            bfrag[c] = *(const v2f*)(drow + 4 * c + 2 * hi);

        v8f c = {};
#pragma unroll
        for (int k = 0; k < 4; ++k) {
            c = __builtin_amdgcn_wmma_f32_16x16x4_f32(
                    false, afrag[k], false, bfrag[k],
                    (short)0, c, false, false);
        }

#pragma unroll
        for (int v = 0; v < 8; ++v)
            acc[v] += __expf(c[v]) * wt;
    }

    // tail chunk (N % 16 != 0), handled by its owner wave with guards
    if ((N & 15) && (nfull % WAVES) == wave) {
        const int n = nfull * 16 + l16;
        const float wt = (n < N) ? weff[n] : 0.0f;
        const float* __restrict__ drow = data + (size_t)((n < N) ? n : 0) * DIMS;

        v2f bfrag[4];
#pragma unroll
        for (int c = 0; c < 4; ++c)
            bfrag[c] = *(const v2f*)(drow + 4 * c + 2 * hi);

        v8f c = {};
#pragma unroll
        for (int k = 0; k < 4; ++k) {
            c = __builtin_amdgcn_wmma_f32_16x16x4_f32(
                    false, afrag[k], false, bfrag[k],
                    (short)0, c, false, false);
        }
#pragma unroll
        for (int v = 0; v < 8; ++v)
            acc[v] += __expf(c[v]) * wt;
    }

    kde_epilogue(lds, acc, X, out, Q, qbase, tid, wave, l16, hi, sum_w);
}

// ---------------------------------------------------------------------------
// Fallback (no workspace): |d_n|^2 computed inline via cross-half shuffle.
// ---------------------------------------------------------------------------
__global__ __launch_bounds__(BLOCK_THREADS)
void kde_wmma_inline(const float* __restrict__ X,
                     const float* __restrict__ data,
                     const float* __restrict__ w,
                     float* __restrict__ out,
                     int Q, int N)
{
    __shared__ float lds[BLOCK_THREADS];

    const int tid  = threadIdx.x;
    const int wave = tid >> 5;
    const int lane = tid & 31;
    const int l16  = lane & 15;
    const int hi   = lane >> 4;

    float sw = 0.0f;
    for (int i = tid; i < N; i += BLOCK_THREADS) sw += w[i];
    lds[tid] = sw;
    __syncthreads();
    for (int s = BLOCK_THREADS / 2; s > 0; s >>= 1) {
        if (tid < s) lds[tid] += lds[tid + s];
        __syncthreads();
    }
    const float sum_w = lds[0];
    __syncthreads();

    const int qbase = blockIdx.x * 16;
    int m = qbase + l16;
    if (m >= Q) m = Q - 1;
    const float* __restrict__ xrow = X + (size_t)m * DIMS;
    v2f afrag[4];
#pragma unroll
    for (int c = 0; c < 4; ++c)
        afrag[c] = *(const v2f*)(xrow + 4 * c + 2 * hi);

    v8f acc = {};
    const int nchunks = (N + 15) >> 4;

    for (int j = wave; j < nchunks; j += WAVES) {
        const int n = j * 16 + l16;
        float wt;
        const float* __restrict__ drow;
        if (n < N) { wt = w[n]; drow = data + (size_t)n * DIMS; }
        else       { wt = 0.0f; drow = data; }

        v2f bfrag[4];
        float dn2p = 0.0f;
#pragma unroll
        for (int c = 0; c < 4; ++c) {
            v2f b = *(const v2f*)(drow + 4 * c + 2 * hi);
            bfrag[c] = b;
            dn2p += b.x * b.x + b.y * b.y;
        }
        // weight with exp(-0.5*|d|^2) folded in; keeps exp(c[v]) independent
        // of the shuffle so the TRANS ops can overlap the DS round-trip.
        const float dn2 = dn2p + __shfl_xor(dn2p, 16, 32);
        const float wf  = wt * __expf(-0.5f * dn2);

        v8f c = {};
#pragma unroll
        for (int k = 0; k < 4; ++k) {
            c = __builtin_amdgcn_wmma_f32_16x16x4_f32(
                    false, afrag[k], false, bfrag[k],
                    (short)0, c, false, false);
        }
#pragma unroll
        for (int v = 0; v < 8; ++v)
            acc[v] += __expf(c[v]) * wf;
    }

    kde_epilogue(lds, acc, X, out, Q, qbase, tid, wave, l16, hi, sum_w);
}

extern "C" void kernel_launch(void* const* d_in, const int* in_sizes, int n_in,
                              void* d_out, int out_size, void* d_ws, size_t ws_size,
                              hipStream_t stream) {
    (void)n_in; (void)out_size;
    const float* X    = (const float*)d_in[0];   // [Q, 16]
    const float* data = (const float*)d_in[1];   // [N, 16]
    const float* w    = (const float*)d_in[2];   // [N]
    float* out = (float*)d_out;                  // [Q]

    const int Q = in_sizes[0] / DIMS;
    const int N = in_sizes[2];
    const int blocks = (Q + 15) / 16;

    if (ws_size >= (size_t)N * sizeof(float)) {
        float* weff = (float*)d_ws;
        kde_prep_weff<<<(N + 255) / 256, 256, 0, stream>>>(data, w, weff, N);
        kde_wmma_fast<<<blocks, BLOCK_THREADS, 0, stream>>>(X, data, weff, w,
                                                            out, Q, N);
    } else {
        kde_wmma_inline<<<blocks, BLOCK_THREADS, 0, stream>>>(X, data, w,
                                                              out, Q, N);
    }
}